// ST_sampling_signal_84842783965716
// MI455X (gfx1250) — compile-verified
//
#include <hip/hip_runtime.h>

// Problem constants from the reference
#define DT    512     // DELTA_T (window length, floats)
#define KWIN  8       // K windows per (b,c)
#define TLEN  4096    // T

#if __has_builtin(__builtin_amdgcn_global_load_async_to_lds_b32) && \
    __has_builtin(__builtin_amdgcn_global_store_async_from_lds_b128)
#define USE_ASYNC 1
#else
#define USE_ASYNC 0
#endif

// Exact pointee types the builtins expect (per hipcc diagnostics):
//   b32  -> int* (AS1 global, AS3 LDS)
//   b128 -> int __attribute__((vector_size(16)))* (AS1 / AS3)
typedef int vint4 __attribute__((vector_size(16)));
typedef __attribute__((address_space(1))) int   g_i32;
typedef __attribute__((address_space(3))) int   l_i32;
typedef __attribute__((address_space(1))) vint4 g_v4i;
typedef __attribute__((address_space(3))) vint4 l_v4i;

__device__ __forceinline__ void wait_asynccnt0() {
#if __has_builtin(__builtin_amdgcn_s_wait_asynccnt)
  __builtin_amdgcn_s_wait_asynccnt(0);
#else
  asm volatile("s_wait_asynccnt 0" ::: "memory");
#endif
}

// One block per (b,c) row; wave w (of 8) copies window k = w.
__global__ __launch_bounds__(256) void ST_sampling_gather_kernel(
    const float* __restrict__ in,     // [B*C, TLEN]
    const int*   __restrict__ offs,   // [B*C, KWIN]
    float*       __restrict__ out)    // [B*C, KWIN, DT]
{
  const int bc   = blockIdx.x;
  const int wave = threadIdx.x >> 5;   // 0..7 -> window index k
  const int lane = threadIdx.x & 31;

  __shared__ __align__(16) float buf[KWIN * DT];   // 16 KB staging

  const int    off  = offs[bc * KWIN + wave];
  const float* src  = in  + (size_t)bc * TLEN + off;                      // 4B-aligned only
  float*       dst  = out + (size_t)bc * (KWIN * DT) + (size_t)wave * DT; // 2KB-aligned
  float*       lbuf = &buf[wave * DT];

#if USE_ASYNC
  // Per-lane bases; per-issue stride lives in the instruction immediate,
  // which the HW adds to BOTH the global and the LDS address.
  g_i32* gld = (g_i32*)(src + lane);        // +4B per lane
  l_i32* lld = (l_i32*)(lbuf + lane);
  // Global -> LDS: 32 lanes x b32 = one 128B cacheline per issue, 16 issues.
  #define ALOAD(i) __builtin_amdgcn_global_load_async_to_lds_b32(gld, lld, (i)*128, 0)
  ALOAD(0);  ALOAD(1);  ALOAD(2);  ALOAD(3);
  ALOAD(4);  ALOAD(5);  ALOAD(6);  ALOAD(7);
  ALOAD(8);  ALOAD(9);  ALOAD(10); ALOAD(11);
  ALOAD(12); ALOAD(13); ALOAD(14); ALOAD(15);
  #undef ALOAD

  wait_asynccnt0();   // this wave's LDS window is complete

  // LDS -> Global: b128 per lane (both sides 16B-aligned), 4 issues.
  // cpol=1 -> TH=NT: output is written once and never re-read; keep L2 for input.
  g_v4i* gst = (g_v4i*)(dst + lane * 4);    // +16B per lane
  l_v4i* lst = (l_v4i*)(lbuf + lane * 4);
  #define ASTORE(i) __builtin_amdgcn_global_store_async_from_lds_b128(gst, lst, (i)*512, 1)
  ASTORE(0); ASTORE(1); ASTORE(2); ASTORE(3);
  #undef ASTORE
  // Async stores drain at the implicit S_WAIT_IDLE of S_ENDPGM.
#else
  // Fallback: direct VGPR copy (b32 loads, b128 stores).
  #pragma unroll
  for (int i = 0; i < 4; ++i) {
    const int j = (i * 32 + lane) * 4;
    float4 v;
    v.x = src[j + 0];
    v.y = src[j + 1];
    v.z = src[j + 2];
    v.w = src[j + 3];
    *(float4*)(dst + j) = v;
  }
  (void)lbuf;
#endif
}

extern "C" void kernel_launch(void* const* d_in, const int* in_sizes, int n_in,
                              void* d_out, int out_size, void* d_ws, size_t ws_size,
                              hipStream_t stream) {
  const float* in   = (const float*)d_in[0];   // [B, C, T] f32
  const int*   offs = (const int*)d_in[1];     // [B, C, K] i32
  float*       out  = (float*)d_out;           // [B, C, K, DT] f32

  const int BC = in_sizes[1] / KWIN;           // B*C = 32768
  ST_sampling_gather_kernel<<<BC, 256, 0, stream>>>(in, offs, out);
}